// CannyExtractor_85109071937580
// MI455X (gfx1250) — compile-verified
//
#include <hip/hip_runtime.h>
#include <cstdint>
#include <cstddef>

#define NB 16
#define H_ 512
#define W_ 512
#define T_ 32
#define GD 40      // gray region:   tile coords [-4, T+4)
#define BVH 36     // blurV rows:    tile coords [-2, T+2)
#define BVW 40     // blurV cols:    tile coords [-4, T+4)
#define BLD 36     // blur:          tile coords [-2, T+2)
#define MGD 34     // mag:           tile coords [-1, T+1)

typedef __attribute__((ext_vector_type(4))) float v4f;

__device__ __forceinline__ int refl_idx(int i, int n) {
    i = (i < 0) ? -i : i;               // jnp.pad 'reflect': -1->1, -2->2
    i = (i >= n) ? (2 * n - 2 - i) : i; //                    n->n-2, n+1->n-3
    return i;
}
__device__ __forceinline__ int clampi(int v, int lo, int hi) {
    return v < lo ? lo : (v > hi ? hi : v);
}

__global__ __launch_bounds__(256)
void canny_fused_kernel(const float* __restrict__ in, float* __restrict__ out) {
    // 4800 floats of channel staging, later reused for blurV / blur / mag.
    __shared__ float s_ch[3 * GD * GD];
    __shared__ float s_gray[GD * GD];          // dead after stage 2 -> reused for dir
    float* s_blurv = s_ch;                     // 36*40 = 1440 floats
    float* s_blur  = s_ch + 1600;              // 36*36 = 1296 floats
    float* s_mag   = s_ch + 3000;              // 34*34 = 1156 floats
    int*   s_dir   = (int*)s_gray;             // 32*32 direction classes

    const int tid = threadIdx.x;
    const int bx  = blockIdx.x * T_;
    const int by  = blockIdx.y * T_;
    const int b   = blockIdx.z;

    // ---- Stage 0: async-gather 3 channel tiles (reflect indexing) into LDS
    const uint64_t base = (uint64_t)(uintptr_t)in +
                          (uint64_t)b * (uint64_t)(3 * H_ * W_) * 4ull;
    for (int idx = tid; idx < GD * GD; idx += 256) {
        int ly = idx / GD, lx = idx - ly * GD;
        int gy = refl_idx(by - 4 + ly, H_);
        int gx = refl_idx(bx - 4 + lx, W_);
        uint64_t a0 = base + (uint64_t)((gy * W_ + gx) * 4);
        uint32_t l0 = (uint32_t)(uintptr_t)(&s_ch[idx]);
        uint32_t l1 = (uint32_t)(uintptr_t)(&s_ch[GD * GD + idx]);
        uint32_t l2 = (uint32_t)(uintptr_t)(&s_ch[2 * GD * GD + idx]);
        asm volatile("global_load_async_to_lds_b32 %0, %1, off"
                     :: "v"(l0), "v"(a0) : "memory");
        asm volatile("global_load_async_to_lds_b32 %0, %1, off"
                     :: "v"(l1), "v"(a0 + (uint64_t)(H_ * W_ * 4)) : "memory");
        asm volatile("global_load_async_to_lds_b32 %0, %1, off"
                     :: "v"(l2), "v"(a0 + (uint64_t)(2 * H_ * W_ * 4)) : "memory");
    }
    asm volatile("s_wait_asynccnt 0" ::: "memory");
    __syncthreads();

    // ---- Stage 1: grayscale (kornia weights)
    for (int idx = tid; idx < GD * GD; idx += 256) {
        s_gray[idx] = 0.299f * s_ch[idx]
                    + 0.587f * s_ch[GD * GD + idx]
                    + 0.114f * s_ch[2 * GD * GD + idx];
    }
    __syncthreads();   // channel planes dead; s_blurv may overwrite them

    // Gaussian 5-tap, sigma=1, normalized
    const float wg[5] = {0.0544886845f, 0.2442013420f, 0.4026199469f,
                         0.2442013420f, 0.0544886845f};

    // ---- Stage 2: vertical blur on 36x40
    for (int idx = tid; idx < BVH * BVW; idx += 256) {
        int y = idx / BVW, x = idx - y * BVW;
        float acc = 0.f;
#pragma unroll
        for (int e = 0; e < 5; ++e) acc += wg[e] * s_gray[(y + e) * GD + x];
        s_blurv[idx] = acc;
    }
    __syncthreads();   // s_gray dead from here on -> reused as s_dir

    // ---- Stage 3: horizontal blur on 36x36
    for (int idx = tid; idx < BLD * BLD; idx += 256) {
        int y = idx / BLD, x = idx - y * BLD;
        float acc = 0.f;
#pragma unroll
        for (int e = 0; e < 5; ++e) acc += wg[e] * s_blurv[y * BVW + x + e];
        s_blur[idx] = acc;
    }
    __syncthreads();

    // ---- Stage 4: Sobel + magnitude on 34x34 (zero outside image = zero-pad
    // of mag) and NMS direction class for the 32x32 center pixels.
    for (int idx = tid; idx < MGD * MGD; idx += 256) {
        int my = idx / MGD, mx = idx - my * MGD;
        int py = by - 1 + my, px = bx - 1 + mx;
        float m = 0.f;
        if (py >= 0 && py < H_ && px >= 0 && px < W_) {
            int r0 = clampi(py - 1, 0, H_ - 1) - (by - 2);
            int r1 = py - (by - 2);
            int r2 = clampi(py + 1, 0, H_ - 1) - (by - 2);
            int c0 = clampi(px - 1, 0, W_ - 1) - (bx - 2);
            int c1 = px - (bx - 2);
            int c2 = clampi(px + 1, 0, W_ - 1) - (bx - 2);
            float b00 = s_blur[r0 * BLD + c0], b01 = s_blur[r0 * BLD + c1], b02 = s_blur[r0 * BLD + c2];
            float b10 = s_blur[r1 * BLD + c0],                              b12 = s_blur[r1 * BLD + c2];
            float b20 = s_blur[r2 * BLD + c0], b21 = s_blur[r2 * BLD + c1], b22 = s_blur[r2 * BLD + c2];
            float gxv = (b02 - b00) + 2.f * (b12 - b10) + (b22 - b20);
            float gyv = (b20 - b00) + 2.f * (b21 - b01) + (b22 - b02);
            m = sqrtf(gxv * gxv + gyv * gyv + 1e-6f);

            // Octant of round(atan2/45deg) mod 8 without atan2; the +/- offset
            // pair is symmetric in the NMS min(), so 4 axis classes suffice.
            // Ties match jnp.round (half-to-even lands on the axis bins).
            if (my >= 1 && my <= T_ && mx >= 1 && mx <= T_) {
                float axv = fabsf(gxv), ayv = fabsf(gyv);
                int cls;
                if (ayv <= 0.41421356f * axv)          cls = 0;  // (0,+-1)
                else if (ayv >= 2.41421356f * axv)     cls = 1;  // (+-1,0)
                else if ((gxv >= 0.f) == (gyv >= 0.f)) cls = 2;  // (+-1,+-1)
                else                                   cls = 3;  // (+-1,-+1)
                s_dir[(my - 1) * T_ + (mx - 1)] = cls;
            }
        }
        s_mag[idx] = m;
    }
    __syncthreads();

    // ---- Stage 5: NMS + clamp; 4 consecutive pixels/thread, b128 NT stores
    {
        const int r     = tid >> 3;         // row 0..31
        const int cbase = (tid & 7) * 4;    // col base 0,4,...,28
        const int py    = by + r;
        float vout[4];
#pragma unroll
        for (int j = 0; j < 4; ++j) {
            int ox  = cbase + j;
            int cls = s_dir[r * T_ + ox];
            int dy  = (cls != 0) ? 1 : 0;
            int dx  = (cls == 1) ? 0 : ((cls == 3) ? -1 : 1);
            float mc = s_mag[(r + 1) * MGD + (ox + 1)];
            float mp = s_mag[(r + 1 + dy) * MGD + (ox + 1 + dx)];
            float mn = s_mag[(r + 1 - dy) * MGD + (ox + 1 - dx)];
            vout[j] = (fminf(mc - mp, mc - mn) > 0.f) ? fminf(mc, 1.f) : 0.f;
        }
        v4f v4 = {vout[0], vout[1], vout[2], vout[3]};
        size_t o = ((size_t)(b * 3) * H_ + (size_t)py) * W_ + (size_t)(bx + cbase);
        __builtin_nontemporal_store(v4, (v4f*)(out + o));
        __builtin_nontemporal_store(v4, (v4f*)(out + o + (size_t)H_ * W_));
        __builtin_nontemporal_store(v4, (v4f*)(out + o + 2 * (size_t)H_ * W_));
    }
}

extern "C" void kernel_launch(void* const* d_in, const int* in_sizes, int n_in,
                              void* d_out, int out_size, void* d_ws, size_t ws_size,
                              hipStream_t stream) {
    (void)in_sizes; (void)n_in; (void)out_size; (void)d_ws; (void)ws_size;
    const float* x = (const float*)d_in[0];
    float* out = (float*)d_out;
    dim3 grid(W_ / T_, H_ / T_, NB);   // 16 x 16 x 16 = 4096 blocks
    canny_fused_kernel<<<grid, 256, 0, stream>>>(x, out);
}